// gin_66632122630729
// MI455X (gfx1250) — compile-verified
//
#include <hip/hip_runtime.h>
#include <hip/hip_bf16.h>
#include <stddef.h>

// ---------------------------------------------------------------------------
// GINE GNN on MI455X (gfx1250, wave32, WMMA).
//   H = 256, N = 50000, E = 800000, 3 GINE layers + 2-layer head.
// Roofline: ~76 GFLOP fp32 GEMM (WMMA f32 16x16x4) vs ~5 GB edge scatter
// traffic (~0.22 ms @ 23.3 TB/s) -> balanced; keep GEMM in fp32 matrix cores
// and make the edge path pure coalesced global loads + fp32 atomics.
// v4: unconditional speculative prefetch (no branch in the k-loop) and
// explicit 4x unroll of the k-loop to restore the deep software pipeline
// (round-1 codegen preloaded 4 k-steps of B in s_clause groups).
// ---------------------------------------------------------------------------

typedef float v2f __attribute__((ext_vector_type(2)));
typedef float v4f __attribute__((ext_vector_type(4)));
typedef float v8f __attribute__((ext_vector_type(8)));

#define HDIM   256
#define NNODES 50000
#define NEDGES 800000
#define MB     64          // node rows per block in the GEMM kernel
#define LDS_S  264         // padded row stride (floats) to spread LDS banks

// ---- global-address-space access helpers (emit global_*, not flat_*) ------
__device__ __forceinline__ float gld(const float* p) {
    return *(const __attribute__((address_space(1))) float*)p;
}
__device__ __forceinline__ v4f gld4(const float* p) {
    return *(const __attribute__((address_space(1))) v4f*)p;
}
__device__ __forceinline__ void gst4(float* p, v4f v) {
    *(__attribute__((address_space(1))) v4f*)p = v;
}
__device__ __forceinline__ long long gldll(const long long* p) {
    return *(const __attribute__((address_space(1))) long long*)p;
}
__device__ __forceinline__ void gatomic_add_f32(float* p, float v) {
    __hip_atomic_fetch_add((__attribute__((address_space(1))) float*)p, v,
                           __ATOMIC_RELAXED, __HIP_MEMORY_SCOPE_AGENT);
}

// ---------------------------------------------------------------------------
// zero the aggregation buffer (128-bit stores)
// ---------------------------------------------------------------------------
__global__ __launch_bounds__(256) void zero_kernel(float* __restrict__ p, int n4) {
    int i = blockIdx.x * 256 + threadIdx.x;
    if (i < n4) gst4(p + (size_t)i * 4, (v4f){0.f, 0.f, 0.f, 0.f});
}

// ---------------------------------------------------------------------------
// Edge kernel: msg = ReLU(x[src] + edge_attr @ We + be); agg[dst] += msg
// 64 threads per edge, 4 H-elements (v4f) per thread, 4 edges per block.
// We is [3][H]; projection is 3 FMAs per element (weights stay in L2/WGP$).
// ---------------------------------------------------------------------------
__global__ __launch_bounds__(256) void edge_msg_kernel(
    const float* __restrict__ x,
    const long long* __restrict__ ei,   // [2][E] int64
    const float* __restrict__ ea,       // [E][3]
    const float* __restrict__ We,       // [3][H]  (this layer)
    const float* __restrict__ be,       // [H]     (this layer)
    float* __restrict__ agg,            // [N][H]
    int E)
{
    const int e  = blockIdx.x * 4 + (threadIdx.x >> 6);
    const int h0 = (threadIdx.x & 63) << 2;
    if (e >= E) return;

    const int src = (int)gldll(ei + e);
    const int dst = (int)gldll(ei + (size_t)E + e);

    const float a0 = gld(ea + (size_t)e * 3 + 0);
    const float a1 = gld(ea + (size_t)e * 3 + 1);
    const float a2 = gld(ea + (size_t)e * 3 + 2);

    const v4f w0 = gld4(We + 0 * HDIM + h0);
    const v4f w1 = gld4(We + 1 * HDIM + h0);
    const v4f w2 = gld4(We + 2 * HDIM + h0);
    const v4f bb = gld4(be + h0);
    const v4f xv = gld4(x + (size_t)src * HDIM + h0);

    v4f m = xv + bb + a0 * w0 + a1 * w1 + a2 * w2;
    m.x = fmaxf(m.x, 0.f);
    m.y = fmaxf(m.y, 0.f);
    m.z = fmaxf(m.z, 0.f);
    m.w = fmaxf(m.w, 0.f);

    float* ap = &agg[(size_t)dst * HDIM + h0];
    gatomic_add_f32(ap + 0, m.x);      // global_atomic_add_f32, no return
    gatomic_add_f32(ap + 1, m.y);
    gatomic_add_f32(ap + 2, m.z);
    gatomic_add_f32(ap + 3, m.w);
}

// ---------------------------------------------------------------------------
// Fused multi-linear kernel:  u = xin (+ agg);  for li < nlin:
//   u = (li < nlin-1 ? ReLU : id)(u @ W[li] + b[li])   (double-buffered LDS)
// epilogue: out = residual ? ReLU(u) + xin : u
//
// Block: 256 threads = 8 wave32. Each block owns MB=64 node rows x 256 cols.
// Each wave owns 2 N-tiles of 16 cols; per N-tile it accumulates 4 M-tiles
// (16x16 each, 8 VGPRs) over K=256 in steps of 4 with v_wmma_f32_16x16x4_f32.
//
// Fragment layouts per CDNA5 ISA 7.12.2 (f32):
//   A 16x4 : lane L -> M = L&15;  VGPR pair = K {kk, kk+1}, kk = k + 2*(L>>4)
//   B 4x16 : lane L -> N = L&15;  VGPR pair = rows {kk, kk+1}
//   C 16x16: lane L, VGPR r -> row = r + 8*(L>>4), col = L&15
// ---------------------------------------------------------------------------
__global__ __launch_bounds__(256) void mlp_fused_kernel(
    const float* __restrict__ xin,
    const float* __restrict__ agg,      // may be null
    const float* __restrict__ W0, const float* __restrict__ b0,
    const float* __restrict__ W1, const float* __restrict__ b1,
    const float* __restrict__ W2, const float* __restrict__ b2,
    int nlin, int residual,
    float* __restrict__ out, int nrows)
{
    __shared__ float bufA[MB * LDS_S];
    __shared__ float bufB[MB * LDS_S];

    const int tid  = threadIdx.x;
    const int row0 = blockIdx.x * MB;

    // ---- load u = x (+ agg) into LDS, zero-pad rows past nrows ----
    for (int idx = tid; idx < MB * (HDIM / 4); idx += 256) {
        const int r = idx >> 6;            // 64 v4f per row
        const int c = (idx & 63) << 2;
        const int gr = row0 + r;
        v4f v = {0.f, 0.f, 0.f, 0.f};
        if (gr < nrows) {
            v = gld4(xin + (size_t)gr * HDIM + c);
            if (agg) v += gld4(agg + (size_t)gr * HDIM + c);
        }
        *(v4f*)&bufA[r * LDS_S + c] = v;
    }
    __syncthreads();

    float* pu = bufA;
    float* ph = bufB;

    const int lane = tid & 31;
    const int wave = tid >> 5;
    const int hl   = lane >> 4;    // lane half (0/1)
    const int nc   = lane & 15;    // col within tile; also M row for A frag

    for (int li = 0; li < nlin; ++li) {
        const float* __restrict__ W    = (li == 0) ? W0 : (li == 1) ? W1 : W2;
        const float* __restrict__ bias = (li == 0) ? b0 : (li == 1) ? b1 : b2;
        const bool doRelu = (li < nlin - 1);

#pragma unroll
        for (int nt = 0; nt < 2; ++nt) {
            const int n0 = (wave * 2 + nt) * 16;

            v8f acc[4] = {};           // 4 M-tiles x 8 VGPRs

#pragma unroll 4
            for (int k = 0; k < HDIM; k += 4) {
                const int kk = k + 2 * hl;
                // speculative prefetch of the W panel 4 k-steps ahead
                // (global_prefetch_b8; unconditional -> no branch in loop,
                //  past-end prefetches are speculative and dropped safely)
                __builtin_prefetch(&W[(size_t)(kk + 16) * HDIM + n0 + nc], 0, 3);
                // B fragment: rows kk, kk+1 of W, cols n0..n0+15 (global loads)
                v2f b;
                b.x = gld(W + (size_t)kk * HDIM + n0 + nc);
                b.y = gld(W + (size_t)(kk + 1) * HDIM + n0 + nc);
#pragma unroll
                for (int mt = 0; mt < 4; ++mt) {
                    // A fragment: u[mt*16 + (lane&15)][kk..kk+1] (8B LDS load)
                    const v2f a = *(const v2f*)&pu[(mt * 16 + nc) * LDS_S + kk];
                    acc[mt] = __builtin_amdgcn_wmma_f32_16x16x4_f32(
                        false, a, false, b, (short)0, acc[mt], false, false);
                }
            }

            const float bv = gld(bias + n0 + nc);
#pragma unroll
            for (int mt = 0; mt < 4; ++mt) {
#pragma unroll
                for (int r = 0; r < 8; ++r) {
                    float v = acc[mt][r] + bv;
                    if (doRelu) v = fmaxf(v, 0.f);
                    ph[(mt * 16 + r + 8 * hl) * LDS_S + n0 + nc] = v;
                }
            }
        }
        __syncthreads();
        float* t = pu; pu = ph; ph = t;   // result of this linear is in pu
    }

    // ---- epilogue: out = residual ? ReLU(u) + xin : u ----
    for (int idx = tid; idx < MB * (HDIM / 4); idx += 256) {
        const int r = idx >> 6;
        const int c = (idx & 63) << 2;
        const int gr = row0 + r;
        if (gr < nrows) {
            v4f h = *(const v4f*)&pu[r * LDS_S + c];
            if (residual) {
                const v4f xv = gld4(xin + (size_t)gr * HDIM + c);
                h.x = fmaxf(h.x, 0.f) + xv.x;
                h.y = fmaxf(h.y, 0.f) + xv.y;
                h.z = fmaxf(h.z, 0.f) + xv.z;
                h.w = fmaxf(h.w, 0.f) + xv.w;
            }
            gst4(out + (size_t)gr * HDIM + c, h);
        }
    }
}

// ---------------------------------------------------------------------------
// Host-side orchestration (graph-capture safe: only kernel launches on stream)
// ---------------------------------------------------------------------------
extern "C" void kernel_launch(void* const* d_in, const int* in_sizes, int n_in,
                              void* d_out, int out_size, void* d_ws, size_t ws_size,
                              hipStream_t stream) {
    const float*      x   = (const float*)d_in[0];
    const long long*  ei  = (const long long*)d_in[1];   // int64 [2][E]
    const float*      ea  = (const float*)d_in[2];       // [E][3]
    const float*      We  = (const float*)d_in[3];       // [3][3][H]
    const float*      be  = (const float*)d_in[4];       // [3][H]
    const float*      Wm  = (const float*)d_in[5];       // [3][3][H][H]
    const float*      bm  = (const float*)d_in[6];       // [3][3][H]
    const float*      pW1 = (const float*)d_in[7];
    const float*      pb1 = (const float*)d_in[8];
    const float*      pW2 = (const float*)d_in[9];
    const float*      pb2 = (const float*)d_in[10];
    float*            out = (float*)d_out;

    const int N = NNODES, E = NEDGES;
    const size_t NH = (size_t)N * HDIM;

    float* ws  = (float*)d_ws;
    float* xA  = ws;             // [N][H]
    float* xB  = ws + NH;        // [N][H]
    float* agg = ws + 2 * NH;    // [N][H]

    const int n4 = (int)(NH / 4);
    dim3 blk(256);
    const int gemmGrid = (N + MB - 1) / MB;
    const int edgeGrid = (E + 3) / 4;
    const int zeroGrid = (n4 + 255) / 256;

    const float* xcur = x;
    float* nexts[3] = { xA, xB, xA };

    for (int l = 0; l < 3; ++l) {
        zero_kernel<<<zeroGrid, blk, 0, stream>>>(agg, n4);
        edge_msg_kernel<<<edgeGrid, blk, 0, stream>>>(
            xcur, ei, ea,
            We + (size_t)l * 3 * HDIM,
            be + (size_t)l * HDIM,
            agg, E);
        const float* Wl = Wm + (size_t)l * 3 * HDIM * HDIM;
        const float* bl = bm + (size_t)l * 3 * HDIM;
        mlp_fused_kernel<<<gemmGrid, blk, 0, stream>>>(
            xcur, agg,
            Wl + 0 * HDIM * HDIM, bl + 0 * HDIM,
            Wl + 1 * HDIM * HDIM, bl + 1 * HDIM,
            Wl + 2 * HDIM * HDIM, bl + 2 * HDIM,
            /*nlin=*/3, /*residual=*/1, nexts[l], N);
        xcur = nexts[l];
    }

    // projection head: Linear -> ReLU -> Linear (no residual)
    mlp_fused_kernel<<<gemmGrid, blk, 0, stream>>>(
        xcur, nullptr,
        pW1, pb1, pW2, pb2, nullptr, nullptr,
        /*nlin=*/2, /*residual=*/0, out, N);
}